// FMFMNeuron_34033320854099
// MI455X (gfx1250) — compile-verified
//
#include <hip/hip_runtime.h>
#include <stdint.h>

// LIF scan: spk[t,b] from spike_seq[T,B,2], W=[0.7,0.6], beta=0.95, thr=1.0
//   cur   = w0*x0 + w1*x1
//   reset = (mem_prev > 1)
//   mem   = 0.95*mem_prev + cur - reset
//   spk   = (mem > 1)
// Memory-bound stream (384 MB @ 23.3 TB/s ~= 16us floor). Pipeline via
// gfx1250 async global->LDS loads (ASYNCcnt), double-buffered per chunk.

#define T_STEPS 512
#define BATCH   65536
#define BPT     2                  // batch columns per thread (one float4/step)
#define BLOCK   128
#define CHUNK   16                 // timesteps per staged chunk
#define NCHUNK  (T_STEPS / CHUNK)  // 32

typedef __attribute__((ext_vector_type(2))) float v2f;
typedef __attribute__((ext_vector_type(4))) float v4f;

// LDS byte address of a __shared__ object (generic -> addrspace(3) -> u32).
__device__ __forceinline__ uint32_t lds_bytes(void* p) {
  return (uint32_t)(uintptr_t)(__attribute__((address_space(3))) char*)p;
}

// One 16B async copy: global -> LDS, tracked by ASYNCcnt (no VGPR return).
__device__ __forceinline__ void async_copy_b128(uint32_t lds_addr, const void* gaddr) {
  asm volatile("global_load_async_to_lds_b128 %0, %1, off"
               :: "v"(lds_addr), "v"(gaddr)
               : "memory");
}

// Waits carry a memory clobber: they are the fence that makes the async LDS
// writes (invisible to the compiler) observable to subsequent LDS reads.
#define WAIT_ASYNCCNT_LE_CHUNK() asm volatile("s_wait_asynccnt 0x10" ::: "memory")
#define WAIT_ASYNCCNT_ZERO()     asm volatile("s_wait_asynccnt 0x0"  ::: "memory")

__global__ __launch_bounds__(BLOCK, 1)
void lif_scan_kernel(const float* __restrict__ x,   // [T, B, 2]
                     const float* __restrict__ W,   // [1, 2]
                     float* __restrict__ out) {     // [T, B, 1]
  __shared__ v4f stage[2][CHUNK][BLOCK];            // 64 KB double buffer

  const int tid = threadIdx.x;
  const int b0  = (blockIdx.x * BLOCK + tid) * BPT; // first batch column
  // float4 view: element (t, b0..b0+1, 0..1) = gin4[t*(B/2) + b0/2]
  const v4f* gin = (const v4f*)x + (b0 >> 1);
  const size_t tstride4 = BATCH / 2;                // float4 per timestep

  const float w0 = W[0];
  const float w1 = W[1];

  // Prologue: stage chunk 0.
  #pragma unroll
  for (int k = 0; k < CHUNK; ++k)
    async_copy_b128(lds_bytes(&stage[0][k][tid]), gin + (size_t)k * tstride4);

  float mem0 = 0.0f, mem1 = 0.0f;

  for (int c = 0; c < NCHUNK; ++c) {
    const int buf = c & 1;

    if (c + 1 < NCHUNK) {
      // Prefetch chunk c+1 into the other buffer, then wait for chunk c
      // (loads complete in order: <=CHUNK outstanding => chunk c landed).
      const v4f* g = gin + (size_t)(c + 1) * CHUNK * tstride4;
      #pragma unroll
      for (int k = 0; k < CHUNK; ++k)
        async_copy_b128(lds_bytes(&stage[buf ^ 1][k][tid]), g + (size_t)k * tstride4);
      WAIT_ASYNCCNT_LE_CHUNK();
    } else {
      WAIT_ASYNCCNT_ZERO();
    }

    #pragma unroll
    for (int k = 0; k < CHUNK; ++k) {
      const v4f xv = stage[buf][k][tid];            // ds_load_b128
      // Two independent batch columns per thread.
      const float cur0 = fmaf(w0, xv.x, w1 * xv.y);
      const float cur1 = fmaf(w0, xv.z, w1 * xv.w);
      const float r0 = (mem0 > 1.0f) ? 1.0f : 0.0f;
      const float r1 = (mem1 > 1.0f) ? 1.0f : 0.0f;
      mem0 = fmaf(0.95f, mem0, cur0) - r0;
      mem1 = fmaf(0.95f, mem1, cur1) - r1;
      v2f s;
      s.x = (mem0 > 1.0f) ? 1.0f : 0.0f;
      s.y = (mem1 > 1.0f) ? 1.0f : 0.0f;
      const size_t t = (size_t)c * CHUNK + k;
      // Streaming output, no reuse: non-temporal 8B coalesced store.
      __builtin_nontemporal_store(s, (v2f*)(out + t * BATCH + b0));
    }
  }
}

extern "C" void kernel_launch(void* const* d_in, const int* in_sizes, int n_in,
                              void* d_out, int out_size, void* d_ws, size_t ws_size,
                              hipStream_t stream) {
  const float* spikes = (const float*)d_in[0]; // [512, 65536, 2] f32
  const float* W      = (const float*)d_in[1]; // [1, 2] f32
  float* out          = (float*)d_out;         // [512, 65536, 1] f32
  (void)in_sizes; (void)n_in; (void)out_size; (void)d_ws; (void)ws_size;

  const int threads = (BATCH / BPT);           // 32768
  dim3 block(BLOCK);
  dim3 grid(threads / BLOCK);                  // 256 blocks
  lif_scan_kernel<<<grid, block, 0, stream>>>(spikes, W, out);
}